// VariableSelectionNetwork_26276609917582
// MI455X (gfx1250) — compile-verified
//
#include <hip/hip_runtime.h>
#include <hip/hip_bf16.h>

// ---------------------------------------------------------------------------
// VariableSelectionNetwork, exact-math-simplified:
//   scorer layernorm is over a size-1 axis => S == sc_ln_b exactly =>
//   softmax weights are data-independent => the 34-GFLOP scorer GRN is dead.
// Remaining work: mixture GEMM + post-GRN, all on V_WMMA_F32_16X16X4_F32.
// x tile staged via global_load_async_to_lds_b32 (ASYNCcnt path).
// ---------------------------------------------------------------------------

typedef __attribute__((ext_vector_type(2))) float v2f;
typedef __attribute__((ext_vector_type(8))) float v8f;

#define BT_ROWS   8192     // B*T
#define FDIM      32
#define DDIM      256
#define HDIM      256
#define MBLK      64       // rows per block
#define LDS_STRIDE 260     // 256 + 4 pad: kills stride-256 bank conflicts
#define XS_STRIDE 34       // even (float2-aligned A frags), 2*17m mod 64 distinct

// ---------------------------------------------------------------------------
// prep0: softmax over sc_ln_b (F=32), fold into projection weights.
//   PW[f][d]  = W[f] * proj_w[f][d]      (already k-major for WMMA B)
//   pbmix[d]  = sum_f W[f] * proj_b[f][d]
// ---------------------------------------------------------------------------
__global__ void vsn_prep0(const float* __restrict__ sc_ln_b,
                          const float* __restrict__ proj_w,
                          const float* __restrict__ proj_b,
                          float* __restrict__ PW,
                          float* __restrict__ pbmix) {
  __shared__ float Wf[FDIM];
  const int tid = threadIdx.x;
  if (tid < FDIM) {
    float mx = -3.4e38f;
    for (int f = 0; f < FDIM; ++f) mx = fmaxf(mx, sc_ln_b[f]);
    float den = 0.f;
    for (int f = 0; f < FDIM; ++f) den += __expf(sc_ln_b[f] - mx);
    Wf[tid] = __expf(sc_ln_b[tid] - mx) / den;
  }
  __syncthreads();
  for (int idx = tid; idx < FDIM * DDIM; idx += 256)
    PW[idx] = Wf[idx >> 8] * proj_w[idx];
  {
    const int c = tid;  // 0..255
    float acc = 0.f;
    for (int f = 0; f < FDIM; ++f)
      acc = fmaf(Wf[f], proj_b[f * DDIM + c], acc);
    pbmix[c] = acc;
  }
}

// ---------------------------------------------------------------------------
// prep1: transpose the three post-GRN weights to k-major B[k][n] layout
// so WMMA B-fragment loads are lane-coalesced.  out[n] = sum_k in[n][k]*x[k].
// ---------------------------------------------------------------------------
__global__ void vsn_prep1(const float* __restrict__ fc1w,   // [H][D]
                          const float* __restrict__ fc2w,   // [D][H]
                          const float* __restrict__ gatew,  // [D][D]
                          float* __restrict__ w1t,
                          float* __restrict__ w2t,
                          float* __restrict__ wgt) {
  const int idx = blockIdx.x * blockDim.x + threadIdx.x;  // 0..65535
  const int n = idx >> 8;
  const int k = idx & 255;
  w1t[k * 256 + n] = fc1w[idx];
  w2t[k * 256 + n] = fc2w[idx];
  wgt[k * 256 + n] = gatew[idx];
}

// ---------------------------------------------------------------------------
// One WMMA GEMM phase: dst[64][256] = act( src[64][KDIM] @ Bt + bias )
// Bt is k-major [KDIM][256].  Each wave owns 2 N-tiles; for each it keeps
// FOUR 16x16 accumulators (all M-tiles) live so every B fragment is reused 4x.
// MODE 0: elu     MODE 1: identity     MODE 2: g=sigmoid(v); y=g*src+(1-g)*mix
// ---------------------------------------------------------------------------
template <int MODE, int KDIM, int SSTR>
__device__ __forceinline__ void gemm_phase(const float* __restrict__ src,
                                           const float* __restrict__ Bt,
                                           const float* __restrict__ bias,
                                           float* __restrict__ dst,
                                           const float* __restrict__ mixbuf) {
  const int tid  = threadIdx.x;
  const int wave = tid >> 5;
  const int lane = tid & 31;
  const int half = lane >> 4;   // 0: lanes 0-15, 1: lanes 16-31
  const int ln   = lane & 15;

  for (int nti = 0; nti < 2; ++nti) {
    const int n0 = (wave * 2 + nti) << 4;
    const float bv = bias[n0 + ln];
    v8f c[4];
#pragma unroll
    for (int mt = 0; mt < 4; ++mt)
      c[mt] = (v8f){bv, bv, bv, bv, bv, bv, bv, bv};

    // A frag: lane = row M (both halves), VGPR0/1 = K = k0+2*half / +1
    const float* ap = src + ln * SSTR + 2 * half;     // 8B aligned (SSTR even)
    // B frag: lane = col N (both halves), rows K = k0+2*half / +1
    const float* bp = Bt + 2 * half * DDIM + (n0 + ln);

#pragma unroll 4
    for (int k0 = 0; k0 < KDIM; k0 += 4) {
      v2f b = {bp[k0 * DDIM], bp[k0 * DDIM + DDIM]};
#pragma unroll
      for (int mt = 0; mt < 4; ++mt) {
        v2f a = *(const v2f*)(ap + mt * 16 * SSTR + k0);
        c[mt] = __builtin_amdgcn_wmma_f32_16x16x4_f32(
            /*neg_a=*/false, a, /*neg_b=*/false, b,
            /*c_mod=*/(short)0, c[mt], /*reuse_a=*/false, /*reuse_b=*/false);
      }
    }

#pragma unroll
    for (int mt = 0; mt < 4; ++mt) {
      const int mb = mt * 16 + 8 * half;  // C/D: VGPR r -> M = r + 8*half
      const int n  = n0 + ln;
#pragma unroll
      for (int r = 0; r < 8; ++r) {
        const int m = mb + r;
        float v = c[mt][r];
        if (MODE == 0) v = (v > 0.f) ? v : (__expf(v) - 1.f);     // elu
        if (MODE == 2) {
          const float g = 1.f / (1.f + __expf(-v));               // sigmoid
          v = g * src[m * SSTR + n] + (1.f - g) * mixbuf[m * SSTR + n];
        }
        dst[m * LDS_STRIDE + n] = v;
      }
    }
  }
}

// ---------------------------------------------------------------------------
// Fused main kernel: 64 rows/block. mix -> (fc1+elu) -> fc2 -> gate+combine
// -> layernorm(D=256) -> out.  Everything between stays in LDS.
// ---------------------------------------------------------------------------
__global__ __launch_bounds__(256) void vsn_main(
    const float* __restrict__ x,       // [8192][32]
    const float* __restrict__ PW,      // [32][256] k-major
    const float* __restrict__ pbmix,   // [256]
    const float* __restrict__ w1t, const float* __restrict__ b1,
    const float* __restrict__ w2t, const float* __restrict__ b2,
    const float* __restrict__ wgt, const float* __restrict__ gb,
    const float* __restrict__ lng, const float* __restrict__ lnb,
    float* __restrict__ out) {        // [8192][256]
  extern __shared__ float smem[];
  float* mixb = smem;                          // 64*260
  float* h1   = mixb + MBLK * LDS_STRIDE;      // 64*260
  float* h2   = h1   + MBLK * LDS_STRIDE;      // 64*260
  float* xs   = h2   + MBLK * LDS_STRIDE;      // 64*34 (reused for LN stats)

  const int tid     = threadIdx.x;
  const int rowBase = blockIdx.x * MBLK;

  // Nudge L2-resident weights toward the WGP (global_prefetch_b8).
  __builtin_prefetch(w1t + tid * 64, 0, 0);
  __builtin_prefetch(w2t + tid * 64, 0, 0);
  __builtin_prefetch(wgt + tid * 64, 0, 0);

  // ---- stage x tile: async global -> LDS (ASYNCcnt path) ----
  // Low 32 bits of a flat LDS pointer == LDS byte address (aperture rule).
#pragma unroll
  for (int i = 0; i < 8; ++i) {
    const int idx = i * 256 + tid;             // 0..2047
    const int r = idx >> 5, f = idx & 31;
    const unsigned long long gp =
        (unsigned long long)(uintptr_t)(x + (size_t)(rowBase + r) * FDIM + f);
    const unsigned lo = (unsigned)(uintptr_t)(xs + r * XS_STRIDE + f);
    asm volatile("global_load_async_to_lds_b32 %0, %1, off"
                 :: "v"(lo), "v"(gp) : "memory");
  }
  asm volatile("s_wait_asynccnt 0x0" ::: "memory");
  __syncthreads();

  // ---- fused pipeline on the fp32 matrix pipe ----
  gemm_phase<1, FDIM, XS_STRIDE>(xs, PW, pbmix, mixb, nullptr);   // mix
  __syncthreads();
  gemm_phase<0, DDIM, LDS_STRIDE>(mixb, w1t, b1, h1, nullptr);    // elu(fc1)
  __syncthreads();
  gemm_phase<1, DDIM, LDS_STRIDE>(h1, w2t, b2, h2, nullptr);      // fc2
  __syncthreads();
  gemm_phase<2, DDIM, LDS_STRIDE>(h2, wgt, gb, h1, mixb);         // gate+mix
  __syncthreads();

  // ---- layernorm over D=256 per row ----
  float* y2    = h1;
  float* stats = xs;  // 64 rows * (4 partial sums + 4 partial sumsq)
  {
    const int r = tid >> 2, q = tid & 3;
    float s = 0.f, s2 = 0.f;
#pragma unroll 16
    for (int d = q * 64; d < q * 64 + 64; ++d) {
      const float v = y2[r * LDS_STRIDE + d];
      s += v;
      s2 = fmaf(v, v, s2);
    }
    stats[r * 8 + q]     = s;
    stats[r * 8 + 4 + q] = s2;
  }
  __syncthreads();
  for (int idx = tid; idx < MBLK * DDIM; idx += 256) {
    const int r = idx >> 8, d = idx & 255;
    const float s  = stats[r*8+0] + stats[r*8+1] + stats[r*8+2] + stats[r*8+3];
    const float s2 = stats[r*8+4] + stats[r*8+5] + stats[r*8+6] + stats[r*8+7];
    const float mean = s * (1.f / 256.f);
    const float var  = s2 * (1.f / 256.f) - mean * mean;
    const float inv  = rsqrtf(var + 1e-5f);
    const float v    = (y2[r * LDS_STRIDE + d] - mean) * inv;
    // streamed 8MB output, never re-read: non-temporal store
    __builtin_nontemporal_store(v * lng[d] + lnb[d],
                                &out[(size_t)(rowBase + r) * DDIM + d]);
  }
}

// ---------------------------------------------------------------------------
extern "C" void kernel_launch(void* const* d_in, const int* in_sizes, int n_in,
                              void* d_out, int out_size, void* d_ws, size_t ws_size,
                              hipStream_t stream) {
  const float* x        = (const float*)d_in[0];
  const float* proj_w   = (const float*)d_in[1];
  const float* proj_b   = (const float*)d_in[2];
  const float* sc_ln_b  = (const float*)d_in[12];
  const float* fc1w     = (const float*)d_in[13];
  const float* fc1b     = (const float*)d_in[14];
  const float* fc2w     = (const float*)d_in[15];
  const float* fc2b     = (const float*)d_in[16];
  const float* gatew    = (const float*)d_in[17];
  const float* gateb    = (const float*)d_in[18];
  const float* lng      = (const float*)d_in[19];
  const float* lnb      = (const float*)d_in[20];

  float* ws    = (float*)d_ws;
  float* PW    = ws;                 // 32*256
  float* pbmix = PW + 32 * 256;      // 256
  float* w1t   = pbmix + 256;        // 256*256
  float* w2t   = w1t + 256 * 256;    // 256*256
  float* wgt   = w2t + 256 * 256;    // 256*256

  const size_t smem =
      (size_t)(3 * MBLK * LDS_STRIDE + MBLK * XS_STRIDE) * sizeof(float);
  (void)hipFuncSetAttribute((const void*)vsn_main,
                            hipFuncAttributeMaxDynamicSharedMemorySize, (int)smem);

  vsn_prep0<<<1, 256, 0, stream>>>(sc_ln_b, proj_w, proj_b, PW, pbmix);
  vsn_prep1<<<256, 256, 0, stream>>>(fc1w, fc2w, gatew, w1t, w2t, wgt);
  vsn_main<<<BT_ROWS / MBLK, 256, smem, stream>>>(x, PW, pbmix, w1t, fc1b,
                                                  w2t, fc2b, wgt, gateb,
                                                  lng, lnb, (float*)d_out);
}